// Seq2SeqWithAttention_65008624992600
// MI455X (gfx1250) — compile-verified
//
#include <hip/hip_runtime.h>

typedef __attribute__((ext_vector_type(2))) float v2f;
typedef __attribute__((ext_vector_type(8))) float v8f;

#define B_  32
#define S_  256
#define T_  64
#define E_  256
#define H_  512
#define H3_ 1536
#define EH_ 768
#define V_  32000

__device__ __forceinline__ v8f wmma4(v2f a, v2f b, v8f c)
{
    return __builtin_amdgcn_wmma_f32_16x16x4_f32(
        /*neg_a=*/false, a, /*neg_b=*/false, b,
        /*c_mod=*/(short)0, c, /*reuse_a=*/false, /*reuse_b=*/false);
}

// ---------------------------------------------------------------------------
// Small fp32 WMMA GEMM: one wave = one 16x16 tile (max wave-parallelism for
// the tiny sequential per-step GEMMs).  C[M,N] = A[M,K] @ W[N,K]^T (+bias).
// A-frag (16x4):  lanes 0-15: M=lane, K=k0,k0+1 ; lanes 16-31: K=k0+2,k0+3
// B-frag (4x16):  lanes 0-15: N=lane, K=k0,k0+1 ; lanes 16-31: K=k0+2,k0+3
// C/D: VGPR r -> (M = tileM + r + 8*halfwave, N = tileN + lane%16)
// ---------------------------------------------------------------------------
__global__ void gemm_wmma_f32(const float* __restrict__ A, int lda,
                              const float* __restrict__ W, int ldw,
                              const float* __restrict__ bias,
                              float* __restrict__ C, int ldc,
                              int tilesN, int K)
{
    const int lane = threadIdx.x & 31;
    const int flat = blockIdx.x * (blockDim.x >> 5) + (threadIdx.x >> 5);
    const int tm = (flat / tilesN) << 4;
    const int tn = (flat % tilesN) << 4;
    const int half = lane >> 4;
    const int l    = lane & 15;

    const float* __restrict__ arow = A + (long)(tm + l) * lda + 2 * half;
    const float* __restrict__ wrow = W + (long)(tn + l) * ldw + 2 * half;

    v8f acc = {};
#pragma unroll 8
    for (int k = 0; k < K; k += 4) {
        v2f a, b;
        a.x = arow[k]; a.y = arow[k + 1];
        b.x = wrow[k]; b.y = wrow[k + 1];
        acc = wmma4(a, b, acc);
    }

    const int col = tn + l;
    const float bv = bias ? bias[col] : 0.0f;
    const int rbase = tm + (half ? 8 : 0);
#pragma unroll
    for (int r = 0; r < 8; ++r)
        C[(long)(rbase + r) * ldc + col] = acc[r] + bv;
}

// ---------------------------------------------------------------------------
// Big register-blocked fp32 WMMA GEMM: one wave = 32x64 macro-tile
// (2 A-frags x 4 B-frags -> 8 WMMA / K-step, 8 f32x8 accumulators).
// Raises arithmetic intensity 4 -> ~10.7 FLOP/B of L2 fragment traffic.
// Requires M%32==0, N%64==0, K%4==0, (M/32)*(N/64)%8==0.
// ---------------------------------------------------------------------------
__global__ void gemm_wmma_f32_blk(const float* __restrict__ A, int lda,
                                  const float* __restrict__ W, int ldw,
                                  const float* __restrict__ bias,
                                  float* __restrict__ C, int ldc,
                                  int tilesN, int K)   // tilesN = N/64
{
    const int lane = threadIdx.x & 31;
    const int flat = blockIdx.x * (blockDim.x >> 5) + (threadIdx.x >> 5);
    const int tm = (flat / tilesN) << 5;   // 32 rows
    const int tn = (flat % tilesN) << 6;   // 64 cols
    const int half = lane >> 4;
    const int l    = lane & 15;

    const float* __restrict__ a0 = A + (long)(tm + l)      * lda + 2 * half;
    const float* __restrict__ a1 = A + (long)(tm + 16 + l) * lda + 2 * half;
    const float* __restrict__ w0 = W + (long)(tn + l)      * ldw + 2 * half;
    const float* __restrict__ w1 = w0 + 16 * (long)ldw;
    const float* __restrict__ w2 = w0 + 32 * (long)ldw;
    const float* __restrict__ w3 = w0 + 48 * (long)ldw;

    v8f c00 = {}, c01 = {}, c02 = {}, c03 = {};
    v8f c10 = {}, c11 = {}, c12 = {}, c13 = {};

#pragma unroll 2
    for (int k = 0; k < K; k += 4) {
        v2f fa0, fa1, fb0, fb1, fb2, fb3;
        fa0.x = a0[k]; fa0.y = a0[k + 1];
        fa1.x = a1[k]; fa1.y = a1[k + 1];
        fb0.x = w0[k]; fb0.y = w0[k + 1];
        fb1.x = w1[k]; fb1.y = w1[k + 1];
        fb2.x = w2[k]; fb2.y = w2[k + 1];
        fb3.x = w3[k]; fb3.y = w3[k + 1];
        c00 = wmma4(fa0, fb0, c00);
        c01 = wmma4(fa0, fb1, c01);
        c02 = wmma4(fa0, fb2, c02);
        c03 = wmma4(fa0, fb3, c03);
        c10 = wmma4(fa1, fb0, c10);
        c11 = wmma4(fa1, fb1, c11);
        c12 = wmma4(fa1, fb2, c12);
        c13 = wmma4(fa1, fb3, c13);
    }

    const float bv0 = bias ? bias[tn + l]      : 0.0f;
    const float bv1 = bias ? bias[tn + 16 + l] : 0.0f;
    const float bv2 = bias ? bias[tn + 32 + l] : 0.0f;
    const float bv3 = bias ? bias[tn + 48 + l] : 0.0f;
    const int r0 = tm + (half ? 8 : 0);
    const int r1 = r0 + 16;
#pragma unroll
    for (int r = 0; r < 8; ++r) {
        float* p0 = C + (long)(r0 + r) * ldc + tn + l;
        float* p1 = C + (long)(r1 + r) * ldc + tn + l;
        p0[0]  = c00[r] + bv0;  p0[16] = c01[r] + bv1;
        p0[32] = c02[r] + bv2;  p0[48] = c03[r] + bv3;
        p1[0]  = c10[r] + bv0;  p1[16] = c11[r] + bv1;
        p1[32] = c12[r] + bv2;  p1[48] = c13[r] + bv3;
    }
}

// ---------------------------------------------------------------------------
__global__ void zero_f32(float* p, int n)
{
    int i = blockIdx.x * blockDim.x + threadIdx.x;
    if (i < n) p[i] = 0.0f;
}

// out[row][0:256] = emb[idx[row]][0:256]   (E = 256, one thread per element)
__global__ void gather_emb256(const int* __restrict__ idx,
                              const float* __restrict__ emb,
                              float* __restrict__ out)
{
    int i = blockIdx.x * blockDim.x + threadIdx.x;
    int row = i >> 8;
    int e   = i & 255;
    out[i] = emb[(long)idx[row] * E_ + e];
}

// ---------------------------------------------------------------------------
// GRU gate fusion. gi row b at gi + b*giStride (includes bih), optional gi2
// (ctx part, stride H3), gh[B,3H] includes bhh. h updated in place; h2 also
// written to out + b*outStride. Gate order [r,z,n]: n = tanh(inn + r*hn),
// h2 = (1-z)*n + z*h
// ---------------------------------------------------------------------------
__global__ void gru_elem(const float* __restrict__ gi, long giStride,
                         const float* __restrict__ gi2,
                         const float* __restrict__ gh,
                         float* __restrict__ h,
                         float* __restrict__ out, long outStride)
{
    int i = blockIdx.x * blockDim.x + threadIdx.x;   // B_*H_ threads
    int b = i >> 9;
    int j = i & (H_ - 1);

    const float* gib = gi + (long)b * giStride;
    const float* ghb = gh + (long)b * H3_;

    float ir = gib[j], iz = gib[H_ + j], inn = gib[2 * H_ + j];
    if (gi2) {
        const float* g2 = gi2 + (long)b * H3_;
        ir += g2[j]; iz += g2[H_ + j]; inn += g2[2 * H_ + j];
    }
    float hr = ghb[j], hz = ghb[H_ + j], hn = ghb[2 * H_ + j];

    float r = 1.0f / (1.0f + __expf(-(ir + hr)));
    float z = 1.0f / (1.0f + __expf(-(iz + hz)));
    float n = tanhf(inn + r * hn);
    float h2 = (1.0f - z) * n + z * h[(long)b * H_ + j];

    h[(long)b * H_ + j] = h2;
    if (out) out[(long)b * outStride + j] = h2;
}

// ---------------------------------------------------------------------------
// scores[b][s] = sum_j tanh(enc_proj[b][s][j] + w1h[b][j]) * v[j]
// one wave32 per (b,s)
// ---------------------------------------------------------------------------
__global__ void attn_scores(const float* __restrict__ enc_proj,
                            const float* __restrict__ w1h,
                            const float* __restrict__ v,
                            float* __restrict__ scores)
{
    int wid  = (blockIdx.x * blockDim.x + threadIdx.x) >> 5;  // 0..B*S-1
    int lane = threadIdx.x & 31;
    int b = wid >> 8;
    int s = wid & (S_ - 1);

    const float* ep = enc_proj + ((long)b * S_ + s) * H_;
    const float* wh = w1h + (long)b * H_;

    float acc = 0.0f;
#pragma unroll 4
    for (int j = lane; j < H_; j += 32)
        acc += tanhf(ep[j] + wh[j]) * v[j];

#pragma unroll
    for (int off = 16; off > 0; off >>= 1)
        acc += __shfl_down(acc, off, 32);

    if (lane == 0) scores[wid] = acc;
}

// in-place softmax over S=256, one block (256 thr) per batch row
__global__ void softmax256(float* __restrict__ scores)
{
    __shared__ float red[256];
    int b = blockIdx.x, s = threadIdx.x;
    float x = scores[b * S_ + s];

    red[s] = x; __syncthreads();
#pragma unroll
    for (int off = 128; off > 0; off >>= 1) {
        if (s < off) red[s] = fmaxf(red[s], red[s + off]);
        __syncthreads();
    }
    float m = red[0]; __syncthreads();

    float e = __expf(x - m);
    red[s] = e; __syncthreads();
#pragma unroll
    for (int off = 128; off > 0; off >>= 1) {
        if (s < off) red[s] += red[s + off];
        __syncthreads();
    }
    scores[b * S_ + s] = e / red[0];
}

// ctx[b][j] = sum_s w[b][s] * enc_outs[b][s][j]
__global__ void attn_ctx(const float* __restrict__ w,
                         const float* __restrict__ enc_outs,
                         float* __restrict__ ctx)
{
    int i = blockIdx.x * blockDim.x + threadIdx.x;   // B_*H_
    int b = i >> 9;
    int j = i & (H_ - 1);
    const float* wb = w + b * S_;
    const float* eb = enc_outs + (long)b * S_ * H_ + j;
    float acc = 0.0f;
#pragma unroll 4
    for (int s = 0; s < S_; ++s)
        acc += wb[s] * eb[(long)s * H_];
    ctx[i] = acc;
}

// ---------------------------------------------------------------------------
extern "C" void kernel_launch(void* const* d_in, const int* in_sizes, int n_in,
                              void* d_out, int out_size, void* d_ws, size_t ws_size,
                              hipStream_t stream)
{
    (void)in_sizes; (void)n_in; (void)out_size; (void)ws_size;

    const int*   src     = (const int*)  d_in[0];
    const int*   tgt     = (const int*)  d_in[1];
    const float* enc_emb = (const float*)d_in[2];
    const float* enc_Wih = (const float*)d_in[3];
    const float* enc_Whh = (const float*)d_in[4];
    const float* enc_bih = (const float*)d_in[5];
    const float* enc_bhh = (const float*)d_in[6];
    const float* dec_emb = (const float*)d_in[7];
    const float* attn_W1 = (const float*)d_in[8];
    const float* attn_b1 = (const float*)d_in[9];
    const float* attn_W2 = (const float*)d_in[10];
    const float* attn_b2 = (const float*)d_in[11];
    const float* attn_v  = (const float*)d_in[12];
    const float* dec_Wih = (const float*)d_in[13];
    const float* dec_Whh = (const float*)d_in[14];
    const float* dec_bih = (const float*)d_in[15];
    const float* dec_bhh = (const float*)d_in[16];
    const float* fc_W    = (const float*)d_in[17];
    const float* fc_b    = (const float*)d_in[18];
    float* out = (float*)d_out;

    // ---- workspace carve-up (floats) ----
    float* wsp = (float*)d_ws;
    size_t o = 0;
    auto alloc = [&](size_t n) { float* r = wsp + o; o += n; return r; };
    float* srcE    = alloc((size_t)B_ * S_ * E_);    // [B,S,E]
    float* GiEnc   = alloc((size_t)B_ * S_ * H3_);   // [B,S,3H]  (x@Wih.T + bih)
    float* encOuts = alloc((size_t)B_ * S_ * H_);    // [B,S,H]
    float* encProj = alloc((size_t)B_ * S_ * H_);    // [B,S,H]
    float* tgtE    = alloc((size_t)B_ * T_ * E_);    // [B,T,E]
    float* GiDec   = alloc((size_t)B_ * T_ * H3_);   // [B,T,3H]  (x part + bih)
    float* h       = alloc((size_t)B_ * H_);         // hidden state
    float* gh      = alloc((size_t)B_ * H3_);        // h@Whh.T + bhh
    float* w1h     = alloc((size_t)B_ * H_);         // h@W1.T + b1
    float* scores  = alloc((size_t)B_ * S_);         // scores / softmax weights
    float* ctx     = alloc((size_t)B_ * H_);         // attention context
    float* giCtx   = alloc((size_t)B_ * H3_);        // ctx@Wih_ctx.T
    float* h2all   = alloc((size_t)B_ * T_ * H_);    // [B,T,H] decoder states

    // small: one 16x16 tile per wave (max parallelism for recurrent steps)
    auto gemm_small = [&](const float* A, int lda, const float* W, int ldw,
                          const float* bias, float* C, int ldc,
                          int M, int N, int K) {
        int tiles = (M / 16) * (N / 16);
        gemm_wmma_f32<<<dim3(tiles / 8), dim3(256), 0, stream>>>(
            A, lda, W, ldw, bias, C, ldc, N / 16, K);
    };
    // big: 32x64 register-blocked tile per wave (L2-traffic optimized)
    auto gemm_big = [&](const float* A, int lda, const float* W, int ldw,
                        const float* bias, float* C, int ldc,
                        int M, int N, int K) {
        int tiles = (M / 32) * (N / 64);
        gemm_wmma_f32_blk<<<dim3(tiles / 8), dim3(256), 0, stream>>>(
            A, lda, W, ldw, bias, C, ldc, N / 64, K);
    };

    // ---- init hidden state ----
    zero_f32<<<dim3((B_ * H_) / 256), dim3(256), 0, stream>>>(h, B_ * H_);

    // ---- encoder: gather embeddings, precompute input projections ----
    gather_emb256<<<dim3((B_ * S_ * E_) / 256), dim3(256), 0, stream>>>(src, enc_emb, srcE);
    gemm_big(srcE, E_, enc_Wih, E_, enc_bih, GiEnc, H3_, B_ * S_, H3_, E_);

    // ---- encoder recurrence ----
    for (int s = 0; s < S_; ++s) {
        gemm_small(h, H_, enc_Whh, H_, enc_bhh, gh, H3_, B_, H3_, H_);
        gru_elem<<<dim3((B_ * H_) / 256), dim3(256), 0, stream>>>(
            GiEnc + (long)s * H3_, (long)S_ * H3_, nullptr, gh, h,
            encOuts + (long)s * H_, (long)S_ * H_);
    }

    // ---- loop-invariant attention projection of encoder outputs ----
    gemm_big(encOuts, H_, attn_W2, H_, attn_b2, encProj, H_, B_ * S_, H_, H_);

    // ---- decoder: gather embeddings, precompute x-part of input proj ----
    gather_emb256<<<dim3((B_ * T_ * E_) / 256), dim3(256), 0, stream>>>(tgt, dec_emb, tgtE);
    // dec_Wih is [3H, E+H]; x multiplies the first E columns (ldw = E+H = 768)
    gemm_big(tgtE, E_, dec_Wih, EH_, dec_bih, GiDec, H3_, B_ * T_, H3_, E_);

    // ---- decoder recurrence ----
    for (int t = 0; t < T_; ++t) {
        gemm_small(h, H_, attn_W1, H_, attn_b1, w1h, H_, B_, H_, H_);
        attn_scores<<<dim3((B_ * S_ * 32) / 256), dim3(256), 0, stream>>>(
            encProj, w1h, attn_v, scores);
        softmax256<<<dim3(B_), dim3(256), 0, stream>>>(scores);
        attn_ctx<<<dim3((B_ * H_) / 256), dim3(256), 0, stream>>>(scores, encOuts, ctx);
        // ctx part of input projection: columns [E, E+H) of dec_Wih
        gemm_small(ctx, H_, dec_Wih + E_, EH_, nullptr, giCtx, H3_, B_, H3_, H_);
        gemm_small(h, H_, dec_Whh, H_, dec_bhh, gh, H3_, B_, H3_, H_);
        gru_elem<<<dim3((B_ * H_) / 256), dim3(256), 0, stream>>>(
            GiDec + (long)t * H3_, (long)T_ * H3_, giCtx, gh, h,
            h2all + (long)t * H_, (long)T_ * H_);
    }

    // ---- vocab projection, hoisted out of the loop: [B*T,512] x [512,32000] ----
    gemm_big(h2all, H_, fc_W, H_, fc_b, out, V_, B_ * T_, V_, H_);
}